// gate_69337952027166
// MI455X (gfx1250) — compile-verified
//
#include <hip/hip_runtime.h>
#include <math.h>

// ---------------------------------------------------------------------------
// Fused router/gate kernel for MI455X (gfx1250, wave32, WMMA).
//
// GEMM view: D[32768 x 32] = x[32768 x 1024] * B[1024 x 32]
//   B cols 0-7  = W_g, 8-15 = W_feat, 16-23 = W_top, 24-31 = 0 (pad)
// computed with V_WMMA_F32_16X16X4_F32; softmax/top-2/sigmoid epilogue done
// in-wave with shuffles. One streaming pass over x (134 MB -> ~5.8 us at the
// 23.3 TB/s HBM floor). Weights are pre-packed once into d_ws so the per-block
// LDS staging is a branch-free b128 copy.
// ---------------------------------------------------------------------------

typedef __attribute__((ext_vector_type(2))) float v2f;
typedef __attribute__((ext_vector_type(8))) float v8f;

#define D_DIM 1024
#define F_DIM 8
#define NCOLS 32
#define KSLICE 256
#define ROWS_PER_BLOCK 128
#define LDS_STRIDE 260   // 260 mod 64 == 4 -> conflict-free b64 B-fragment reads
#define WS_BYTES (NCOLS * D_DIM * sizeof(float))   // 128 KB packed B

// One-shot: pack the three [1024 x 8] weight matrices (row-major over d) into
// transposed B' [32 cols][1024 k] in workspace. Runs once per launch; tiny.
__global__ __launch_bounds__(256) void pack_weights_kernel(
    const float* __restrict__ Wg,
    const float* __restrict__ Wf,
    const float* __restrict__ Wt,
    float* __restrict__ wp)
{
    const int idx = blockIdx.x * 256 + threadIdx.x;   // 0 .. 32*1024-1
    const int n = idx >> 10;          // packed column 0..31
    const int k = idx & (D_DIM - 1);  // k 0..1023
    float v = 0.0f;
    if (n < 8)        v = Wg[k * F_DIM + n];
    else if (n < 16)  v = Wf[k * F_DIM + (n - 8)];
    else if (n < 24)  v = Wt[k * F_DIM + (n - 16)];
    wp[n * D_DIM + k] = v;
}

template <bool USE_WS>
__global__ __launch_bounds__(256) void fused_router_kernel(
    const float* __restrict__ x,
    const float* __restrict__ wp,    // packed B' [32][1024] (USE_WS path)
    const float* __restrict__ Wg, const float* __restrict__ bg,
    const float* __restrict__ Wf, const float* __restrict__ bf,
    const float* __restrict__ Wt, const float* __restrict__ bt,
    const float* __restrict__ alpha,
    float* __restrict__ out,
    int rows)
{
    __shared__ __align__(16) float bsm[NCOLS * LDS_STRIDE];  // 33.3 KB of 320 KB WGP LDS

    const int tid  = threadIdx.x;
    const int lane = tid & 31;
    const int hf   = lane >> 4;      // 16-lane half of the wave
    const int l15  = lane & 15;
    const int wave = tid >> 5;       // 0..7

    const size_t rowBase = (size_t)blockIdx.x * ROWS_PER_BLOCK + (size_t)wave * 16;
    const float* xrow = x + (rowBase + (size_t)l15) * D_DIM;

    v8f c0 = {};   // D columns 0-15  (gate | soft-router logits)
    v8f c1 = {};   // D columns 16-31 (top-k logits | pad)

    for (int ko = 0; ko < D_DIM; ko += KSLICE) {
        __syncthreads();
        if (USE_WS) {
            // Branch-free b128 staging: 2048 float4s, 8 per thread.
            for (int i = tid; i < NCOLS * KSLICE / 4; i += 256) {
                const int n   = i >> 6;           // packed column
                const int kk4 = (i & 63) << 2;    // local k (float4 granularity)
                const float4 v = *(const float4*)(wp + n * D_DIM + ko + kk4);
                *(float4*)(bsm + n * LDS_STRIDE + kk4) = v;
            }
        } else {
            // Fallback: gather-pack straight from the raw weights.
            for (int i = tid; i < NCOLS * KSLICE; i += 256) {
                const int n  = i >> 8;
                const int kk = i & (KSLICE - 1);
                const int k  = ko + kk;
                float v = 0.0f;
                if (n < 8)        v = Wg[k * F_DIM + n];
                else if (n < 16)  v = Wf[k * F_DIM + (n - 8)];
                else if (n < 24)  v = Wt[k * F_DIM + (n - 16)];
                bsm[n * LDS_STRIDE + kk] = v;
            }
        }
        __syncthreads();

        if (ko + KSLICE < D_DIM)                       // global_prefetch_b8 of next x slice
            __builtin_prefetch(xrow + ko + KSLICE, 0, 0);

        const float* b0p = bsm + l15 * LDS_STRIDE;          // B cols 0-15
        const float* b1p = bsm + (l15 + 16) * LDS_STRIDE;   // B cols 16-31

        #pragma unroll 4
        for (int kc = 0; kc < KSLICE / 4; ++kc) {
            const int kl = kc * 4 + hf * 2;   // A/B fragment layout: half 0 -> K{0,1}, half 1 -> K{2,3}
            v2f a  = *(const v2f*)(xrow + ko + kl);   // A: 16x4 fp32 tile
            v2f b0 = *(const v2f*)(b0p + kl);         // B: 4x16 tiles from LDS
            v2f b1 = *(const v2f*)(b1p + kl);
            c0 = __builtin_amdgcn_wmma_f32_16x16x4_f32(false, a, false, b0,
                                                       (short)0, c0, false, false);
            c1 = __builtin_amdgcn_wmma_f32_16x16x4_f32(false, a, false, b1,
                                                       (short)0, c1, false, false);
        }
    }

    // ------------------------- epilogue (in-wave) ---------------------------
    // C/D layout: VGPR r holds row (r + 8*hf); lane column = l15 (c0) / 16+l15 (c1).
    // lanes l15 0-7: gate logits (c0) + top-k logits (c1); lanes 8-15: soft logits.
    const float ab    = 1.0f / (1.0f + __expf(-alpha[0]));          // sigmoid(alpha)
    const float biasA = (l15 < 8) ? bg[l15] : bf[l15 - 8];
    const float biasB = (l15 < 8) ? bt[l15] : -__builtin_inff();    // mask idle lanes for top-2

    #pragma unroll
    for (int r = 0; r < 8; ++r) {
        float v = c0[r] + biasA;

        // softmax across each 8-lane group (valid result lives in lanes 8-15)
        float m = v;
        m = fmaxf(m, __shfl_xor(m, 1));
        m = fmaxf(m, __shfl_xor(m, 2));
        m = fmaxf(m, __shfl_xor(m, 4));
        float e = __expf(v - m);
        float s = e;
        s += __shfl_xor(s, 1);
        s += __shfl_xor(s, 2);
        s += __shfl_xor(s, 4);
        const float soft   = e / s;
        const float soft_f = __shfl_xor(soft, 8);         // feature f's soft value -> lane f
        const float gate   = 1.0f / (1.0f + __expf(-v));  // valid in lanes 0-7

        // top-2 reduction over the 8 top-k logits (lanes 0-7 of the group)
        float t  = c1[r] + biasB;
        float t1 = t, t2 = -__builtin_inff();
        #pragma unroll
        for (int msk = 1; msk <= 4; msk <<= 1) {
            const float o1 = __shfl_xor(t1, msk);
            const float o2 = __shfl_xor(t2, msk);
            const float hi = fmaxf(t1, o1);
            const float lo = fminf(t1, o1);
            t2 = fmaxf(lo, fmaxf(t2, o2));
            t1 = hi;
        }
        // softmax over {t1, t2}: p1 = sigmoid(t1 - t2)
        const float p1 = 1.0f / (1.0f + __expf(t2 - t1));
        const float p2 = 1.0f - p1;
        const float sv = (t == t1) ? p1 : ((t == t2) ? p2 : 0.0f);

        float contrib = gate * (ab * sv + (1.0f - ab) * soft_f);
        contrib += __shfl_xor(contrib, 1);
        contrib += __shfl_xor(contrib, 2);
        contrib += __shfl_xor(contrib, 4);

        const size_t row = rowBase + (size_t)r + (size_t)(8 * hf);
        if (l15 == 0 && row < (size_t)rows)
            out[row] = contrib;
    }
}

extern "C" void kernel_launch(void* const* d_in, const int* in_sizes, int n_in,
                              void* d_out, int out_size, void* d_ws, size_t ws_size,
                              hipStream_t stream) {
    (void)n_in; (void)out_size;
    const float* x     = (const float*)d_in[0];
    const float* Wg    = (const float*)d_in[1];
    const float* bg    = (const float*)d_in[2];
    const float* Wf    = (const float*)d_in[3];
    const float* bf    = (const float*)d_in[4];
    const float* Wt    = (const float*)d_in[5];
    const float* bt    = (const float*)d_in[6];
    const float* alpha = (const float*)d_in[7];
    // d_in[8] = expand (k=2), hardcoded in the top-2 epilogue.

    const int rows = in_sizes[0] / D_DIM;                           // 32768
    const int grid = (rows + ROWS_PER_BLOCK - 1) / ROWS_PER_BLOCK;  // 256 blocks

    if (ws_size >= WS_BYTES) {
        float* wp = (float*)d_ws;
        pack_weights_kernel<<<(NCOLS * D_DIM) / 256, 256, 0, stream>>>(Wg, Wf, Wt, wp);
        fused_router_kernel<true><<<grid, 256, 0, stream>>>(
            x, wp, Wg, bg, Wf, bf, Wt, bt, alpha, (float*)d_out, rows);
    } else {
        fused_router_kernel<false><<<grid, 256, 0, stream>>>(
            x, nullptr, Wg, bg, Wf, bf, Wt, bt, alpha, (float*)d_out, rows);
    }
}